// BasicRangeProjection_35751307772112
// MI455X (gfx1250) — compile-verified
//
#include <hip/hip_runtime.h>
#include <hip/hip_bf16.h>

// ---------------------------------------------------------------------------
// BasicRangeProjection for MI455X (gfx1250, wave32).
// Two-pass pixel-winner scatter:
//   pass 0: init winner-index buffer (d_ws) to -1
//   pass 1: per-point projection, racy store of point index per pixel
//           (nondeterministic winner == XLA scatter-set with dup indices)
//   pass 2: per-pixel gather + recompute features; all 7 channel-plane
//           stores are wave32-coalesced and non-temporal (write-once output).
// Fallback (tiny ws): zero-init + direct per-point scatter.
// ---------------------------------------------------------------------------

namespace rp {

constexpr int   Wd   = 2048;
constexpr int   Hd   = 64;
constexpr int   Bd   = 8;
constexpr int   Cd   = 7;               // x,y,z,r,theta,phi,intensity
constexpr int   NPIX = Bd * Hd * Wd;    // 1,048,576 pixels

constexpr float PI_F      = 3.14159265358979323846f;
constexpr float DEG_F     = 0.017453292519943295f;          // pi/180
constexpr float INV_HRNG  = 1.0f / (2.0f * PI_F);           // 1/(HFOV span)
constexpr float VLO       = -3.0f * DEG_F;                  // V_FOV[0]
constexpr float INV_VRNG  = 1.0f / (28.0f * DEG_F);         // 1/(VFOV span)

__device__ __forceinline__ void project(float x, float y, float z,
                                        float& r, float& theta, float& phi,
                                        int& u, int& v, bool& inb) {
    r     = sqrtf(x * x + y * y + z * z);
    theta = -atan2f(y, x);
    phi   = -asinf(z / fmaxf(r, 1e-5f));
    float un = (theta + PI_F) * INV_HRNG;
    float vn = (phi - VLO) * INV_VRNG;
    inb = (un >= 0.0f) & (un < 1.0f) & (vn >= 0.0f) & (vn < 1.0f);
    u = (int)(un * (float)Wd);
    v = (int)(vn * (float)Hd);
}

// ---- pass 0: winner buffer = -1 (int4 stores, 16B per lane) ---------------
__global__ __launch_bounds__(256) void init_idx(int* __restrict__ idx) {
    int t = blockIdx.x * blockDim.x + threadIdx.x;   // NPIX/4 threads
    if (t < NPIX / 4) {
        ((int4*)idx)[t] = make_int4(-1, -1, -1, -1);
    }
}

// ---- pass 1: per-point pixel-winner scatter -------------------------------
__global__ __launch_bounds__(256) void scatter_idx(const float* __restrict__ pts,
                                                   int* __restrict__ idx, int n) {
    int i = blockIdx.x * blockDim.x + threadIdx.x;
    if (i >= n) return;
    const float* p = pts + (size_t)i * 5;
    float bsf = p[0], x = p[1], y = p[2], z = p[3];
    float r, theta, phi; int u, v; bool inb;
    project(x, y, z, r, theta, phi, u, v, inb);
    if (inb) {
        int b   = (int)bsf;                            // in [0, B)
        int pix = (b * Hd + v) * Wd + u;
        idx[pix] = i;                                  // racy: arbitrary winner
    }
}

// ---- pass 2: per-pixel gather; coalesced NT channel-plane stores ----------
__global__ __launch_bounds__(256) void gather_write(const float* __restrict__ pts,
                                                    const int* __restrict__ idx,
                                                    float* __restrict__ out) {
    int t = blockIdx.x * blockDim.x + threadIdx.x;     // NPIX threads
    if (t >= NPIX) return;
    int u = t & (Wd - 1);                              // consecutive lanes -> consecutive u
    int v = (t >> 11) & (Hd - 1);
    int b = t >> 17;

    int i = idx[t];
    float f[Cd] = {0.f, 0.f, 0.f, 0.f, 0.f, 0.f, 0.f};
    if (i >= 0) {
        const float* p = pts + (size_t)i * 5;
        float x = p[1], y = p[2], z = p[3], inten = p[4];
        float r, theta, phi; int uu, vv; bool inb;
        project(x, y, z, r, theta, phi, uu, vv, inb);
        f[0] = x; f[1] = y; f[2] = z; f[3] = r; f[4] = theta; f[5] = phi; f[6] = inten;
    }
#pragma unroll
    for (int c = 0; c < Cd; ++c) {
        size_t o = (((size_t)(b * Cd + c) * Hd + v) * Wd) + u;  // (B,C,H,W)
        __builtin_nontemporal_store(f[c], out + o);             // TH_STORE_NT stream
    }
}

// ---- fallback: small workspace --------------------------------------------
__global__ __launch_bounds__(256) void zero_out(float* __restrict__ out, int n4) {
    int t = blockIdx.x * blockDim.x + threadIdx.x;
    if (t < n4) {
        ((float4*)out)[t] = make_float4(0.f, 0.f, 0.f, 0.f);
    }
}

__global__ __launch_bounds__(256) void direct_scatter(const float* __restrict__ pts,
                                                      float* __restrict__ out, int n) {
    int i = blockIdx.x * blockDim.x + threadIdx.x;
    if (i >= n) return;
    const float* p = pts + (size_t)i * 5;
    float bsf = p[0], x = p[1], y = p[2], z = p[3], inten = p[4];
    float r, theta, phi; int u, v; bool inb;
    project(x, y, z, r, theta, phi, u, v, inb);
    if (!inb) return;
    int b = (int)bsf;
    float f[Cd] = {x, y, z, r, theta, phi, inten};
#pragma unroll
    for (int c = 0; c < Cd; ++c) {
        size_t o = (((size_t)(b * Cd + c) * Hd + v) * Wd) + u;
        out[o] = f[c];
    }
}

} // namespace rp

extern "C" void kernel_launch(void* const* d_in, const int* in_sizes, int n_in,
                              void* d_out, int out_size, void* d_ws, size_t ws_size,
                              hipStream_t stream) {
    using namespace rp;
    const float* points = (const float*)d_in[0];   // (N, 5) float32
    float*       out    = (float*)d_out;           // (8, 7, 64, 2048) float32
    const int N = in_sizes[0] / 5;

    const int threads = 256;
    const size_t ws_need = (size_t)NPIX * sizeof(int);

    if (ws_size >= ws_need) {
        int* idx = (int*)d_ws;
        init_idx<<<(NPIX / 4 + threads - 1) / threads, threads, 0, stream>>>(idx);
        scatter_idx<<<(N + threads - 1) / threads, threads, 0, stream>>>(points, idx, N);
        gather_write<<<(NPIX + threads - 1) / threads, threads, 0, stream>>>(points, idx, out);
    } else {
        const int n4 = out_size / 4;   // 7,340,032 floats, divisible by 4
        zero_out<<<(n4 + threads - 1) / threads, threads, 0, stream>>>(out, n4);
        direct_scatter<<<(N + threads - 1) / threads, threads, 0, stream>>>(points, out, N);
    }
}